// Encoder_20512763806039
// MI455X (gfx1250) — compile-verified
//
#include <hip/hip_runtime.h>

// ---------------------------------------------------------------------------
// FCOS target encoder for MI455X (gfx1250, wave32).
// - ltrb outer-difference tiles via V_WMMA_F32_16X16X4_F32 (bit-exact fp32,
//   K=0-only operands so the result is independent of B K-row interleave).
// - gt_boxes/classes staged to LDS with global_load_async_to_lds_b32 (+
//   s_wait_asynccnt), the CDNA5 async-copy path.
// - argmin via packed (area_bits<<32 | idx) u64 keys, shfl_xor width-16.
// ---------------------------------------------------------------------------

typedef __attribute__((ext_vector_type(2))) float v2f;
typedef __attribute__((ext_vector_type(8))) float v8f;

#define HW_TOTAL 21824   // 128^2 + 64^2 + 32^2 + 16^2 + 8^2
#define NB       8
#define M_GT     100
#define M_PAD    112     // 7 box tiles of 16
#define NTILES   1364    // HW_TOTAL / 16
#define WPB      8       // waves per block
#define BIGAREA  9999999.0f

__device__ __forceinline__ v8f wmma_f32_4(v2f a, v2f b, v8f c) {
  return __builtin_amdgcn_wmma_f32_16x16x4_f32(
      /*neg_a=*/false, a, /*neg_b=*/false, b,
      /*c_mod=*/(short)0, c, /*reuse_a=*/false, /*reuse_b=*/false);
}

__global__ __launch_bounds__(256)
void fcos_targets_kernel(const float* __restrict__ gt_boxes,   // [B,100,4]
                         const int*   __restrict__ classes,    // [B,100]
                         float* __restrict__ out)               // cls|cnt|reg
{
  __shared__ float sh_gt[M_PAD * 4];
  __shared__ float sh_cx[M_PAD];
  __shared__ float sh_cy[M_PAD];
  __shared__ int   sh_cls[M_PAD];

  const int b = blockIdx.y;
  const int t = threadIdx.x;
  const float* gtb = gt_boxes + b * M_GT * 4;
  const int*   clb = classes  + b * M_GT;

  // ---- stage gt data into LDS via CDNA5 async global->LDS DMA ----
  for (int i = t; i < M_GT * 4; i += 256) {
    unsigned lp = (unsigned)(unsigned long long)&sh_gt[i];
    asm volatile("global_load_async_to_lds_b32 %0, %1, off"
                 :: "v"(lp), "v"(gtb + i) : "memory");
  }
  for (int i = t; i < M_GT; i += 256) {
    unsigned lp = (unsigned)(unsigned long long)&sh_cls[i];
    asm volatile("global_load_async_to_lds_b32 %0, %1, off"
                 :: "v"(lp), "v"(clb + i) : "memory");
  }
  // zero-fill pad boxes 100..111 (plain LDS stores, disjoint region)
  if (t < (M_PAD - M_GT) * 4) sh_gt[M_GT * 4 + t] = 0.0f;
  if (t < (M_PAD - M_GT))     sh_cls[M_GT + t] = 0;
  asm volatile("s_wait_asynccnt 0x0" ::: "memory");
  __syncthreads();
  if (t < M_PAD) {
    sh_cx[t] = (sh_gt[t * 4 + 0] + sh_gt[t * 4 + 2]) * 0.5f;
    sh_cy[t] = (sh_gt[t * 4 + 1] + sh_gt[t * 4 + 3]) * 0.5f;
  }
  __syncthreads();

  const int wave = t >> 5;
  const int lane = t & 31;
  const int tile = blockIdx.x * WPB + wave;
  if (tile >= NTILES) return;

  // ---- level parameters (wave-uniform; 16-loc tiles never cross levels) ----
  const int tl = tile * 16;                 // flat loc within batch
  int lvl_start, log2w, log2s; float lim0, lim1;
  if (tl < 16384)      { lvl_start = 0;     log2w = 7; log2s = 3; lim0 = -1.f;  lim1 = 64.f;     }
  else if (tl < 20480) { lvl_start = 16384; log2w = 6; log2s = 4; lim0 = 64.f;  lim1 = 128.f;    }
  else if (tl < 21504) { lvl_start = 20480; log2w = 5; log2s = 5; lim0 = 128.f; lim1 = 256.f;    }
  else if (tl < 21760) { lvl_start = 21504; log2w = 4; log2s = 6; lim0 = 256.f; lim1 = 512.f;    }
  else                 { lvl_start = 21760; log2w = 3; log2s = 7; lim0 = 512.f; lim1 = 999999.f; }
  const int   half_s = 1 << (log2s - 1);
  const float radius = (float)(1 << log2s) * 1.5f;
  const int   rel    = tl - lvl_start;
  const int   wmask  = (1 << log2w) - 1;

  // ---- A-side coordinates: row m = lane (lanes 0-15), K=0 only ----
  const int  mlane = lane & 15;
  const bool lo    = lane < 16;
  {
    // nothing: coords computed below
  }
  int   pA = rel + mlane;
  float xA = (float)(((pA & wmask) << log2s) + half_s);
  float yA = (float)(((pA >> log2w) << log2s) + half_s);

  v2f aX   = { lo ?  xA  : 0.f, 0.f };
  v2f aY   = { lo ?  yA  : 0.f, 0.f };
  v2f aNX  = { lo ? -xA  : 0.f, 0.f };
  v2f aNY  = { lo ? -yA  : 0.f, 0.f };
  v2f ones = { lo ? 1.0f : 0.f, 0.f };
  v8f zc = {};
  // Broadcast tiles: TX[m,n]=x[m], TY[m,n]=y[m], TNX=-x[m], TNY=-y[m]
  v8f TX  = wmma_f32_4(aX,  ones, zc);
  v8f TY  = wmma_f32_4(aY,  ones, zc);
  v8f TNX = wmma_f32_4(aNX, ones, zc);
  v8f TNY = wmma_f32_4(aNY, ones, zc);

  // per-slot (row) coordinates: slot v -> loc row v + 8*(lane>=16)
  const int half8 = (lane >> 4) << 3;
  float xs_[8], ys_[8];
#pragma unroll
  for (int v = 0; v < 8; ++v) {
    int p = rel + v + half8;
    xs_[v] = (float)(((p & wmask) << log2s) + half_s);
    ys_[v] = (float)(((p >> log2w) << log2s) + half_s);
  }

  unsigned long long key[8];
#pragma unroll
  for (int v = 0; v < 8; ++v) key[v] = ~0ull;
  unsigned pm = 0;

  for (int bt = 0; bt < 7; ++bt) {
    const int box = (bt << 4) + mlane;       // box index for this lane's slots
    float x1 = sh_gt[box * 4 + 0], y1 = sh_gt[box * 4 + 1];
    float x2 = sh_gt[box * 4 + 2], y2 = sh_gt[box * 4 + 3];
    float cx = sh_cx[box], cy = sh_cy[box];
    const bool valid = box < M_GT;

    v2f bNX1 = { lo ? -x1 : 0.f, 0.f };
    v2f bNY1 = { lo ? -y1 : 0.f, 0.f };
    v2f bX2  = { lo ?  x2 : 0.f, 0.f };
    v2f bY2  = { lo ?  y2 : 0.f, 0.f };

    // l = x[m]-x1[n], t = y[m]-y1[n], r = x2[n]-x[m], b = y2[n]-y[m]
    v8f L  = wmma_f32_4(ones, bNX1, TX);
    v8f T  = wmma_f32_4(ones, bNY1, TY);
    v8f R  = wmma_f32_4(ones, bX2,  TNX);
    v8f Bo = wmma_f32_4(ones, bY2,  TNY);

#pragma unroll
    for (int v = 0; v < 8; ++v) {
      float l = L[v], tt = T[v], r = R[v], bb = Bo[v];
      float omin = fminf(fminf(l, tt), fminf(r, bb));
      float omax = fmaxf(fmaxf(l, tt), fmaxf(r, bb));
      float cmax = fmaxf(fabsf(xs_[v] - cx), fabsf(ys_[v] - cy));
      bool maskpos = valid && (omin > 0.f) && (omin > lim0) &&
                     (omax <= lim1) && (cmax < radius);
      float area = __fmul_rn(__fadd_rn(l, r), __fadd_rn(tt, bb));
      float cand = maskpos ? area : BIGAREA;
      unsigned long long ck =
          ((unsigned long long)__float_as_uint(cand) << 32) | (unsigned)box;
      if (ck < key[v]) key[v] = ck;
      pm |= maskpos ? (1u << v) : 0u;
    }
  }

  // ---- reduce over the 16 box-lanes (N dim) per slot ----
#pragma unroll
  for (int v = 0; v < 8; ++v) {
    unsigned long long k = key[v];
#pragma unroll
    for (int m = 1; m < 16; m <<= 1) {
      unsigned long long o = __shfl_xor(k, m, 16);
      k = (o < k) ? o : k;
    }
    key[v] = k;
  }
#pragma unroll
  for (int m = 1; m < 16; m <<= 1)
    pm |= (unsigned)__shfl_xor((int)pm, m, 16);

  // ---- writers: lanes 0-7 emit locs 0-7, lanes 16-23 emit locs 8-15 ----
  if (mlane >= 8) return;
  const int v    = mlane;
  const int locj = mlane + half8;
  unsigned long long kk = key[0];
#pragma unroll
  for (int i = 1; i < 8; ++i) kk = (v == i) ? key[i] : kk;
  const bool pos = ((pm >> v) & 1u) != 0;
  const int bidx = (int)(unsigned)kk;

  int   p = rel + locj;
  float x = (float)(((p & wmask) << log2s) + half_s);
  float y = (float)(((p >> log2w) << log2s) + half_s);

  float clsv = 0.f, cntv = -1.f;
  float l = -1.f, tt = -1.f, r = -1.f, bb = -1.f;
  if (pos) {
    float x1 = sh_gt[bidx * 4 + 0], y1 = sh_gt[bidx * 4 + 1];
    float x2 = sh_gt[bidx * 4 + 2], y2 = sh_gt[bidx * 4 + 3];
    l = x - x1; tt = y - y1; r = x2 - x; bb = y2 - y;
    float lrmin = fminf(l, r),  lrmax = fmaxf(l, r);
    float tbmin = fminf(tt, bb), tbmax = fmaxf(tt, bb);
    float den = __fadd_rn(__fmul_rn(lrmax, tbmax), 1e-10f);
    cntv = sqrtf(__fdiv_rn(__fmul_rn(lrmin, tbmin), den));
    clsv = (float)sh_cls[bidx];
  }

  const int g = b * HW_TOTAL + tl + locj;
  out[g] = clsv;                                   // cls_targets [B,HW,1]
  out[NB * HW_TOTAL + g] = cntv;                   // cnt_targets [B,HW,1]
  float* rg = out + 2 * NB * HW_TOTAL + (size_t)g * 4;
  rg[0] = l; rg[1] = tt; rg[2] = r; rg[3] = bb;    // reg_targets [B,HW,4]
}

extern "C" void kernel_launch(void* const* d_in, const int* in_sizes, int n_in,
                              void* d_out, int out_size, void* d_ws, size_t ws_size,
                              hipStream_t stream) {
  (void)in_sizes; (void)n_in; (void)d_ws; (void)ws_size; (void)out_size;
  const float* gt_boxes = (const float*)d_in[15];   // [8,100,4] f32
  const int*   classes  = (const int*)  d_in[16];   // [8,100] i32
  dim3 grid((NTILES + WPB - 1) / WPB, NB);
  fcos_targets_kernel<<<grid, 256, 0, stream>>>(gt_boxes, classes, (float*)d_out);
}